// LightDLGN2_32770600469235
// MI455X (gfx1250) — compile-verified
//
#include <hip/hip_runtime.h>

// DLGN recurrent logic-gate network for MI455X (gfx1250).
// Gather/VALU-bound workload: state lives in the 320KB WGP LDS; epilogue
// reduction uses V_WMMA_F32_16X16X4_F32 with an all-ones A matrix.

#define BT       16                 // batch tile per workgroup
#define NTHREADS 256
#define NSLOT    (NTHREADS / BT)    // 16 gate slots
#define W0       2048
#define W1       1024
#define W2       384
#define PFEAT    784
#define FBN      256
#define POPN     128
#define NIN0     (PFEAT + FBN)      // 1040
#define STEPSN   8
#define NGATES   (W0 + W1 + W2)     // 3456

typedef float v2f __attribute__((ext_vector_type(2)));
typedef float v8f __attribute__((ext_vector_type(8)));

// ---------------- pre-pass: fold logits -> bilinear coefficients -------------
// f = (1-L)(1-R)w0 + (1-L)R w1 + L(1-R) w2 + L R w3
//   = w0 + (w2-w0) L + (w1-w0) R + (w0-w1-w2+w3) L R
__global__ void dlgn_precomp(const float* __restrict__ lg0,
                             const int*  __restrict__ le0, const int* __restrict__ ri0,
                             const float* __restrict__ lg1,
                             const int*  __restrict__ le1, const int* __restrict__ ri1,
                             const float* __restrict__ lg2,
                             const int*  __restrict__ le2, const int* __restrict__ ri2,
                             int2*  __restrict__ idx, float4* __restrict__ co) {
  int G = blockIdx.x * blockDim.x + threadIdx.x;
  if (G >= NGATES) return;
  const float* lg; const int* li; const int* ri; int g;
  if (G < W0)            { lg = lg0; li = le0; ri = ri0; g = G; }
  else if (G < W0 + W1)  { lg = lg1; li = le1; ri = ri1; g = G - W0; }
  else                   { lg = lg2; li = le2; ri = ri2; g = G - W0 - W1; }
  float w0 = 0.5f + 0.5f * sinf(lg[g * 4 + 0]);
  float w1 = 0.5f + 0.5f * sinf(lg[g * 4 + 1]);
  float w2 = 0.5f + 0.5f * sinf(lg[g * 4 + 2]);
  float w3 = 0.5f + 0.5f * sinf(lg[g * 4 + 3]);
  co[G]  = make_float4(w0, w2 - w0, w1 - w0, w0 - w1 - w2 + w3);
  idx[G] = make_int2(li[g], ri[g]);
}

// ---------------- main kernel ------------------------------------------------
__global__ __launch_bounds__(NTHREADS)
void dlgn_main(const float* __restrict__ x, const float* __restrict__ thr,
               const int2* __restrict__ idx, const float4* __restrict__ co,
               float* __restrict__ out) {
  __shared__ float sh_hin[NIN0 * BT];   //  66.5 KB : layer-0 input (part | fb)
  __shared__ float sh_l0 [W0   * BT];   // 131.0 KB : layer-0 output
  __shared__ float sh_l1 [W1   * BT];   //  65.5 KB : layer-1 output
  __shared__ float sh_pop[POPN * BT];   //   8.0 KB : population counts

  const int t    = threadIdx.x;
  const int bb   = t & (BT - 1);        // batch lane within tile
  const int slot = t / BT;              // gate slot (uniform per half-wave)
  const float* xb = x + (size_t)(blockIdx.x * BT + bb) * PFEAT;

  // p = slot + NSLOT*j  =>  p & 7 == slot & 7 (loop-invariant threshold)
  const float myth = thr[slot & 7];

  // feedback starts at zero
  for (int j = slot; j < FBN; j += NSLOT)
    sh_hin[(PFEAT + j) * BT + bb] = 0.0f;

  for (int s = 0; s < STEPSN; ++s) {
    // ---- encode this step's 784 binary features (pixels 98s .. 98s+97) ----
    const float* xs = xb + s * 98;
    for (int p = slot; p < PFEAT; p += NSLOT) {
      float v = xs[p >> 3];
      sh_hin[p * BT + bb] = (v >= myth) ? 1.0f : 0.0f;
    }
    if (s + 1 < STEPSN) __builtin_prefetch(xs + 98, 0, 0);  // global_prefetch_b8
    __syncthreads();

    // ---- layer 0 : 1040 -> 2048 ----
    for (int g = slot; g < W0; g += NSLOT) {
      int2   lr = idx[g];
      float4 w  = co[g];
      float L = sh_hin[lr.x * BT + bb];
      float R = sh_hin[lr.y * BT + bb];
      sh_l0[g * BT + bb] = w.x + w.y * L + w.z * R + w.w * (L * R);
    }
    __syncthreads();

    // ---- layer 1 : 2048 -> 1024 ----
    for (int g = slot; g < W1; g += NSLOT) {
      int2   lr = idx[W0 + g];
      float4 w  = co[W0 + g];
      float L = sh_l0[lr.x * BT + bb];
      float R = sh_l0[lr.y * BT + bb];
      sh_l1[g * BT + bb] = w.x + w.y * L + w.z * R + w.w * (L * R);
    }
    __syncthreads();

    // ---- layer 2 : 1024 -> 384 (256 feedback | 128 pop) ----
    for (int g = slot; g < W2; g += NSLOT) {
      int2   lr = idx[W0 + W1 + g];
      float4 w  = co[W0 + W1 + g];
      float L = sh_l1[lr.x * BT + bb];
      float R = sh_l1[lr.y * BT + bb];
      float v = w.x + w.y * L + w.z * R + w.w * (L * R);
      if (g < FBN) sh_hin[(PFEAT + g) * BT + bb] = v;
      else         sh_pop[(g - FBN) * BT + bb]  = v;
    }
    __syncthreads();
  }

  // ---- epilogue: logit[col] = sum_{j<128} pop[j][col] via f32 WMMA ----
  // A = all-ones 16x4  =>  D[m][n] = sum_k B[k][n]; exact fp32, one wave,
  // invariant to the K-slot permutation inside B's VGPR layout.
  if (t < 32) {                         // wave 0 only; EXEC all-ones inside
    v8f acc = {};
    v2f ones; ones[0] = 1.0f; ones[1] = 1.0f;
    const int col   = t & 15;
    const int khalf = (t < 16) ? 0 : 2; // lanes 0-15: K rows 0,1; 16-31: 2,3
    #pragma unroll 4
    for (int kc = 0; kc < 32; ++kc) {   // 32 chunks x K=4 = 128 pop rows
      v2f bm;
      bm[0] = sh_pop[(4 * kc + khalf + 0) * BT + col];
      bm[1] = sh_pop[(4 * kc + khalf + 1) * BT + col];
      acc = __builtin_amdgcn_wmma_f32_16x16x4_f32(
          /*neg_a=*/false, ones, /*neg_b=*/false, bm,
          /*c_mod=*/(short)0, acc, /*reuse_a=*/false, /*reuse_b=*/false);
    }
    if (t < 16) {                       // D vgpr0, lanes 0-15 = row M=0
      float logit = acc[0] / 10.0f;     // TAU
      float* ob = out + (size_t)(blockIdx.x * BT + t) * 10;
      #pragma unroll
      for (int c = 0; c < 10; ++c) ob[c] = logit;
    }
  }
}

// ---------------- launcher ---------------------------------------------------
extern "C" void kernel_launch(void* const* d_in, const int* in_sizes, int n_in,
                              void* d_out, int out_size, void* d_ws, size_t ws_size,
                              hipStream_t stream) {
  const float* x   = (const float*)d_in[0];   // (4096,1,28,28)
  const float* thr = (const float*)d_in[1];   // (8,)
  const float* lg0 = (const float*)d_in[2];   // (2048,4)
  const int*   le0 = (const int*)  d_in[3];
  const int*   ri0 = (const int*)  d_in[4];
  const float* lg1 = (const float*)d_in[5];   // (1024,4)
  const int*   le1 = (const int*)  d_in[6];
  const int*   ri1 = (const int*)  d_in[7];
  const float* lg2 = (const float*)d_in[8];   // (384,4)
  const int*   le2 = (const int*)  d_in[9];
  const int*   ri2 = (const int*)  d_in[10];

  int2*   idx = (int2*)d_ws;                          // 3456 * 8B
  float4* co  = (float4*)((char*)d_ws + 32768);       // 3456 * 16B

  dlgn_precomp<<<(NGATES + 255) / 256, 256, 0, stream>>>(
      lg0, le0, ri0, lg1, le1, ri1, lg2, le2, ri2, idx, co);

  dlgn_main<<<4096 / BT, NTHREADS, 0, stream>>>(x, thr, idx, co, (float*)d_out);
}